// KthGate_53523882443557
// MI455X (gfx1250) — compile-verified
//
#include <hip/hip_runtime.h>
#include <hip/hip_bf16.h>
#include <cstdint>

// Problem constants (match reference)
constexpr int S   = 4096;   // tokens
constexpr int Dm  = 4096;   // model dim
constexpr int E   = 64;     // experts
constexpr int CAP = 128;    // capacity
constexpr int KCH = 4;      // k-split chunks
constexpr int KC_LEN = Dm / KCH;       // 1024
constexpr float LB_W = 0.01f;
constexpr float F32_EPS = 1.1920928955078125e-7f;

typedef __attribute__((ext_vector_type(2))) float v2f;
typedef __attribute__((ext_vector_type(8))) float v8f;

// ---------------------------------------------------------------------------
// Kernel 1: logits partials = x @ W^T   (fp32 WMMA 16x16x4, exact fp32)
// grid = (256 row tiles, 4 k-chunks), block = 32 (one wave)
// Each wave: 16 rows x 64 experts, K range [kc*1024, kc*1024+1024)
// A 16x4 f32 layout: lane(0-15)=M, VGPR{0,1}=K{0,1}; lanes 16-31 hold K{2,3}
// B 4x16 f32 layout: lane(0-15)=N, VGPR{0,1}=K{0,1}; lanes 16-31 hold K{2,3}
// -> per lane: one float2 from x row, one float2 from each W row group.
// ---------------------------------------------------------------------------
__global__ __launch_bounds__(32) void moe_gemm_wmma(
    const float* __restrict__ x, const float* __restrict__ Wt,
    float* __restrict__ partials) {
  const int bt = blockIdx.x;          // row tile 0..255
  const int kc = blockIdx.y;          // k chunk  0..3
  const int l  = threadIdx.x;         // 0..31
  const int lm = l & 15;
  const int lh = l >> 4;
  const size_t k0 = (size_t)kc * KC_LEN + 2 * lh;

  const float* pa  = x  + (size_t)(bt * 16 + lm) * Dm + k0;
  const float* pb0 = Wt + (size_t)( 0 + lm) * Dm + k0;
  const float* pb1 = Wt + (size_t)(16 + lm) * Dm + k0;
  const float* pb2 = Wt + (size_t)(32 + lm) * Dm + k0;
  const float* pb3 = Wt + (size_t)(48 + lm) * Dm + k0;

  v8f acc0 = {}, acc1 = {}, acc2 = {}, acc3 = {};

#pragma unroll 8
  for (int i = 0; i < KC_LEN / 4; ++i) {
    const int off = i * 4;
    v2f a  = *reinterpret_cast<const v2f*>(pa  + off);
    v2f b0 = *reinterpret_cast<const v2f*>(pb0 + off);
    v2f b1 = *reinterpret_cast<const v2f*>(pb1 + off);
    v2f b2 = *reinterpret_cast<const v2f*>(pb2 + off);
    v2f b3 = *reinterpret_cast<const v2f*>(pb3 + off);
    acc0 = __builtin_amdgcn_wmma_f32_16x16x4_f32(false, a, false, b0, (short)0, acc0, false, false);
    acc1 = __builtin_amdgcn_wmma_f32_16x16x4_f32(false, a, false, b1, (short)0, acc1, false, false);
    acc2 = __builtin_amdgcn_wmma_f32_16x16x4_f32(false, a, false, b2, (short)0, acc2, false, false);
    acc3 = __builtin_amdgcn_wmma_f32_16x16x4_f32(false, a, false, b3, (short)0, acc3, false, false);
  }

  // C/D layout: VGPR r, lanes 0-15 -> M=r, lanes 16-31 -> M=r+8; N = lane%16
  float* po = partials + (size_t)kc * S * E;
#pragma unroll
  for (int r = 0; r < 8; ++r) {
    const int row = bt * 16 + r + 8 * lh;
    po[(size_t)row * E +  0 + lm] = acc0[r];
    po[(size_t)row * E + 16 + lm] = acc1[r];
    po[(size_t)row * E + 32 + lm] = acc2[r];
    po[(size_t)row * E + 48 + lm] = acc3[r];
  }
}

// ---------------------------------------------------------------------------
// Kernel 2: per-token softmax + top1/top2 selection. One thread per token.
// ---------------------------------------------------------------------------
__global__ __launch_bounds__(256) void moe_softmax_top2(
    const float* __restrict__ partials, const float* __restrict__ bias,
    float* __restrict__ gates, int* __restrict__ idx1, int* __restrict__ idx2,
    float* __restrict__ g1raw, float* __restrict__ g2raw) {
  const int s = blockIdx.x * blockDim.x + threadIdx.x;
  if (s >= S) return;

  float lg[E];
  const float* p = partials + (size_t)s * E;
#pragma unroll
  for (int e = 0; e < E; ++e) {
    lg[e] = p[e] + p[e + (size_t)S * E] + p[e + 2 * (size_t)S * E] +
            p[e + 3 * (size_t)S * E] + bias[e];
  }

  // top-2 scan; strict '>' reproduces argmax-first-occurrence and stable
  // argsort second-place tie semantics.
  float v1 = -3.4e38f, v2 = -3.4e38f;
  int i1 = 0, i2 = 0;
#pragma unroll
  for (int e = 0; e < E; ++e) {
    float v = lg[e];
    if (v > v1) { v2 = v1; i2 = i1; v1 = v; i1 = e; }
    else if (v > v2) { v2 = v; i2 = e; }
  }

  float den = 0.f;
#pragma unroll
  for (int e = 0; e < E; ++e) den += expf(lg[e] - v1);
  const float inv = 1.0f / den;

  float* gr = gates + (size_t)s * E;
#pragma unroll
  for (int e = 0; e < E; ++e) gr[e] = expf(lg[e] - v1) * inv;

  idx1[s] = i1;
  idx2[s] = i2;
  g1raw[s] = inv;                       // exp(0)/den = max gate
  g2raw[s] = expf(v2 - v1) * inv;
}

// ---------------------------------------------------------------------------
// Kernel 3: per-expert block — count1[e], me_sum[e], priority rank loc1,
// prefix-count loc2. All reductions deterministic (LDS trees / fixed loops).
// ---------------------------------------------------------------------------
__global__ __launch_bounds__(256) void moe_rank(
    const int* __restrict__ idx1, const int* __restrict__ idx2,
    const float* __restrict__ g1raw, const float* __restrict__ gates,
    int* __restrict__ loc1s, int* __restrict__ loc2s,
    float* __restrict__ me_sum, int* __restrict__ count1) {
  __shared__ int   s_i1[S];
  __shared__ int   s_i2[S];
  __shared__ float s_mg[S];
  __shared__ float s_rf[256];
  __shared__ int   s_ri[256];

  const int e = blockIdx.x;
  const int t = threadIdx.x;

  for (int s = t; s < S; s += 256) {
    s_i1[s] = idx1[s];
    s_i2[s] = idx2[s];
    s_mg[s] = g1raw[s];                 // max gate == importance key
  }
  __syncthreads();

  // count1[e] and me_sum[e]
  int   cnt = 0;
  float me  = 0.f;
  for (int s = t; s < S; s += 256) {
    cnt += (s_i1[s] == e) ? 1 : 0;
    me  += gates[(size_t)s * E + e];
  }
  s_rf[t] = me; s_ri[t] = cnt;
  __syncthreads();
  for (int w = 128; w > 0; w >>= 1) {
    if (t < w) { s_rf[t] += s_rf[t + w]; s_ri[t] += s_ri[t + w]; }
    __syncthreads();
  }
  const int cnt_e = s_ri[0];
  if (t == 0) { me_sum[e] = s_rf[0]; count1[e] = cnt_e; }

  // loc1: rank among same-expert tokens in (maxgate desc, index asc) order.
  // loc2: strict prefix count in original order + count1[e].
  for (int s = t; s < S; s += 256) {
    if (s_i1[s] == e) {
      const float mv = s_mg[s];
      int c = 0;
      for (int s2 = 0; s2 < S; ++s2) {
        bool higher = (s_mg[s2] > mv) || ((s_mg[s2] == mv) && (s2 < s));
        c += (s_i1[s2] == e && higher) ? 1 : 0;
      }
      loc1s[s] = c;
    }
    if (s_i2[s] == e) {
      int c = 0;
      for (int s2 = 0; s2 < s; ++s2) c += (s_i2[s2] == e) ? 1 : 0;
      loc2s[s] = c + cnt_e;
    }
  }
}

// ---------------------------------------------------------------------------
// Kernel 4: l_aux scalar
// ---------------------------------------------------------------------------
__global__ void moe_laux(const float* __restrict__ me_sum,
                         const int* __restrict__ count1,
                         float* __restrict__ out) {
  if (threadIdx.x == 0 && blockIdx.x == 0) {
    float acc = 0.f;
    for (int e = 0; e < E; ++e) {
      float mev = me_sum[e] / (float)S;
      float cev = (float)count1[e] / (float)S;
      acc += mev * cev;
    }
    out[0] = acc * (float)E * LB_W;
  }
}

// ---------------------------------------------------------------------------
// Kernel 5: write combine + dispatch (zero everywhere, scatter <=2 values).
// One block per token; fully covers d_out[1..] so no separate memset needed.
// ---------------------------------------------------------------------------
__global__ __launch_bounds__(256) void moe_output(
    const int* __restrict__ idx1, const int* __restrict__ idx2,
    const int* __restrict__ loc1s, const int* __restrict__ loc2s,
    const float* __restrict__ g1raw, const float* __restrict__ g2raw,
    float* __restrict__ out) {
  const int s = blockIdx.x;
  const int t = threadIdx.x;

  const int i1 = idx1[s], i2 = idx2[s];
  const int l1 = loc1s[s], l2 = loc2s[s];
  const bool k1 = l1 < CAP;
  const bool k2 = l2 < CAP;
  float g1 = k1 ? g1raw[s] : 0.f;
  float g2 = k2 ? g2raw[s] : 0.f;
  const float den = fmaxf(g1 + g2, F32_EPS);
  g1 /= den; g2 /= den;

  const int p1 = k1 ? (i1 * CAP + l1) : -1;
  const int p2 = k2 ? (i2 * CAP + l2) : -2;

  float* comb = out + 1 + (size_t)s * (E * CAP);
  float* disp = out + 1 + (size_t)S * E * CAP + (size_t)s * (E * CAP);

  for (int p = t; p < E * CAP; p += 256) {
    float v = (p == p1) ? g1 : ((p == p2) ? g2 : 0.f);
    comb[p] = v;
    disp[p] = (v != 0.f) ? 1.f : 0.f;
  }
}

// ---------------------------------------------------------------------------
extern "C" void kernel_launch(void* const* d_in, const int* in_sizes, int n_in,
                              void* d_out, int out_size, void* d_ws, size_t ws_size,
                              hipStream_t stream) {
  const float* x  = (const float*)d_in[0];   // [S, Dm]
  const float* Wt = (const float*)d_in[1];   // [E, Dm]
  const float* b  = (const float*)d_in[2];   // [E]
  float* out = (float*)d_out;

  // Workspace layout
  float* partials = (float*)d_ws;                       // KCH * S * E
  float* gates    = partials + (size_t)KCH * S * E;     // S * E
  int*   idx1     = (int*)(gates + (size_t)S * E);      // S
  int*   idx2     = idx1 + S;                           // S
  float* g1raw    = (float*)(idx2 + S);                 // S
  float* g2raw    = g1raw + S;                          // S
  int*   loc1s    = (int*)(g2raw + S);                  // S
  int*   loc2s    = loc1s + S;                          // S
  float* me_sum   = (float*)(loc2s + S);                // E
  int*   count1   = (int*)(me_sum + E);                 // E

  moe_gemm_wmma<<<dim3(S / 16, KCH), 32, 0, stream>>>(x, Wt, partials);
  moe_softmax_top2<<<S / 256, 256, 0, stream>>>(partials, b, gates, idx1, idx2,
                                                g1raw, g2raw);
  moe_rank<<<E, 256, 0, stream>>>(idx1, idx2, g1raw, gates, loc1s, loc2s,
                                  me_sum, count1);
  moe_laux<<<1, 1, 0, stream>>>(me_sum, count1, out);
  moe_output<<<S, 256, 0, stream>>>(idx1, idx2, loc1s, loc2s, g1raw, g2raw, out);
}